// Imputation_37958920962331
// MI455X (gfx1250) — compile-verified
//
#include <hip/hip_runtime.h>
#include <hip/hip_bf16.h>
#include <stdint.h>

// GraphConv GNN on MI455X (gfx1250, wave32).
// Dense layers use V_WMMA_F32_16X16X4_F32 (exact fp32 matmul, no precision loss).
// Edge scatter uses coalesced global_atomic_add_f32 (the HBM-bound part).

typedef float v2f __attribute__((ext_vector_type(2)));
typedef float v8f __attribute__((ext_vector_type(8)));

#define N_NODES 50000
#define N_EDGES 800000
#define IN_CH   64
#define HID     128

// ---------------------------------------------------------------- utilities
__global__ void zero_kernel(float* __restrict__ p, int n) {
    int i = blockIdx.x * blockDim.x + threadIdx.x;
    if (i < n) p[i] = 0.0f;
}

// counts[dst[e]] += 1
__global__ void count_kernel(const int* __restrict__ dst, float* __restrict__ counts, int nE) {
    int e = blockIdx.x * blockDim.x + threadIdx.x;
    if (e < nE) atomicAdd(&counts[dst[e]], 1.0f);
}

// agg[dst[e]*C + c] += ew[e] * h[src[e]*C + c]; one thread per (edge, channel)
__global__ void scatter_add_kernel(const int* __restrict__ src, const int* __restrict__ dst,
                                   const float* __restrict__ ew, const float* __restrict__ h,
                                   float* __restrict__ agg, int C, int total) {
    int gid = blockIdx.x * blockDim.x + threadIdx.x;
    if (gid >= total) return;
    int e = gid / C;
    int c = gid - e * C;
    int s = src[e];
    int d = dst[e];
    float v = ew[e] * h[(size_t)s * C + c];
    atomicAdd(&agg[(size_t)d * C + c], v);
}

// agg[i*C+c] /= max(counts[i], 1)
__global__ void scale_kernel(float* __restrict__ agg, const float* __restrict__ counts,
                             int C, int total) {
    int gid = blockIdx.x * blockDim.x + threadIdx.x;
    if (gid >= total) return;
    int i = gid / C;
    float d = counts[i];
    d = d > 1.0f ? d : 1.0f;
    agg[gid] = agg[gid] / d;
}

// ---------------------------------------------------------------- WMMA GEMM
// Out[M x O] = relu( A1[M x K1] @ W1[O x K1]^T + bias[O] (+ A2[M x K2] @ W2[O x K2]^T) )
// One wave per 16x16 output tile. fp32 WMMA, K step = 4.
// A layout (16x4 fp32): lane l in [0,16) holds row mt*16+l; VGPR0 = K(k+2*half),
// VGPR1 = K(k+2*half+1). B mirrors with N. C/D: vgpr r, lane l, half h ->
// (M = mt*16 + 8*h + r, N = nt*16 + l).
__global__ void gemm_dual_relu_wmma(const float* __restrict__ A1, const float* __restrict__ W1, int K1,
                                    const float* __restrict__ A2, const float* __restrict__ W2, int K2,
                                    const float* __restrict__ bias, float* __restrict__ Out,
                                    int M, int O) {
    const int wave = threadIdx.x >> 5;
    const int lane = threadIdx.x & 31;
    const int ntO  = O >> 4;                       // 16-wide tiles along O
    const int tile = blockIdx.x * (blockDim.x >> 5) + wave;
    const int mt   = tile / ntO;
    const int nt   = tile - mt * ntO;
    if (mt * 16 >= M) return;                      // wave-uniform: EXEC stays all-ones

    const int half = lane >> 4;                    // 0 or 1: selects K sub-pair
    const int l    = lane & 15;
    const int arow = mt * 16 + l;                  // A row this lane feeds
    const int bcol = nt * 16 + l;                  // output feature this lane holds

    v8f c = {0.f, 0.f, 0.f, 0.f, 0.f, 0.f, 0.f, 0.f};

    {
        const float* __restrict__ ap = A1 + (size_t)arow * K1 + 2 * half;
        const float* __restrict__ wp = W1 + (size_t)bcol * K1 + 2 * half;
        for (int k = 0; k < K1; k += 4) {
            v2f a, b;
            a.x = ap[k];  a.y = ap[k + 1];
            b.x = wp[k];  b.y = wp[k + 1];
            c = __builtin_amdgcn_wmma_f32_16x16x4_f32(
                    false, a, false, b, (short)0, c, false, false);
        }
    }
    if (A2 != nullptr) {
        const float* __restrict__ ap = A2 + (size_t)arow * K2 + 2 * half;
        const float* __restrict__ wp = W2 + (size_t)bcol * K2 + 2 * half;
        for (int k = 0; k < K2; k += 4) {
            v2f a, b;
            a.x = ap[k];  a.y = ap[k + 1];
            b.x = wp[k];  b.y = wp[k + 1];
            c = __builtin_amdgcn_wmma_f32_16x16x4_f32(
                    false, a, false, b, (short)0, c, false, false);
        }
    }

    const float bv = bias[bcol];
    #pragma unroll
    for (int r = 0; r < 8; ++r) {
        int orow = mt * 16 + half * 8 + r;
        float v = c[r] + bv;
        Out[(size_t)orow * O + bcol] = v > 0.0f ? v : 0.0f;
    }
}

// ---------------------------------------------------------------- final 128 -> 1
__global__ void predict_kernel(const float* __restrict__ h, const float* __restrict__ Wp,
                               const float* __restrict__ bp, float* __restrict__ out,
                               int M, int K) {
    int i = blockIdx.x * blockDim.x + threadIdx.x;
    if (i >= M) return;
    const float4* hr = (const float4*)(h + (size_t)i * K);
    const float4* wr = (const float4*)Wp;
    float s = 0.0f;
    for (int k = 0; k < K / 4; ++k) {
        float4 a = hr[k], b = wr[k];
        s += a.x * b.x + a.y * b.y + a.z * b.z + a.w * b.w;
    }
    out[i] = s + bp[0];
}

// ---------------------------------------------------------------- driver
static inline int cdiv(int a, int b) { return (a + b - 1) / b; }

extern "C" void kernel_launch(void* const* d_in, const int* in_sizes, int n_in,
                              void* d_out, int out_size, void* d_ws, size_t ws_size,
                              hipStream_t stream) {
    const float* x       = (const float*)d_in[0];        // N x 64
    const int*   eidx    = (const int*)d_in[1];          // 2 x E
    const float* ew      = (const float*)d_in[2];        // E
    const float* W_rel0  = (const float*)d_in[3];        // 128 x 64
    const float* b_rel0  = (const float*)d_in[4];
    const float* W_root0 = (const float*)d_in[5];        // 128 x 64
    const float* W_ro0   = (const float*)d_in[6];        // 128 x 128
    const float* b_ro0   = (const float*)d_in[7];
    const float* W_rel1  = (const float*)d_in[8];        // 128 x 128
    const float* b_rel1  = (const float*)d_in[9];
    const float* W_root1 = (const float*)d_in[10];       // 128 x 128
    const float* W_ro1   = (const float*)d_in[11];       // 128 x 128
    const float* b_ro1   = (const float*)d_in[12];
    const float* W_prd   = (const float*)d_in[13];       // 1 x 128
    const float* b_prd   = (const float*)d_in[14];
    float* out = (float*)d_out;                          // N x 1

    const int* src = eidx;
    const int* dst = eidx + N_EDGES;

    // Workspace layout (floats): counts | bufA | bufB | bufC   (~77 MB total)
    float* ws     = (float*)d_ws;
    const int NM  = N_NODES * HID;                       // 6.4M floats
    float* counts = ws;
    float* bufA   = ws + 50176;
    float* bufB   = bufA + NM;
    float* bufC   = bufB + NM;

    const int BLK = 256;
    const int gemmWaves = BLK / 32;

    // 1) in-degree counts
    zero_kernel<<<cdiv(N_NODES, BLK), BLK, 0, stream>>>(counts, N_NODES);
    count_kernel<<<cdiv(N_EDGES, BLK), BLK, 0, stream>>>(dst, counts, N_EDGES);

    // 2) conv0 aggregate: agg0 (N x 64) in bufA
    {
        int tot = N_NODES * IN_CH;
        zero_kernel<<<cdiv(tot, BLK), BLK, 0, stream>>>(bufA, tot);
        int etot = N_EDGES * IN_CH;
        scatter_add_kernel<<<cdiv(etot, BLK), BLK, 0, stream>>>(src, dst, ew, x, bufA, IN_CH, etot);
        scale_kernel<<<cdiv(tot, BLK), BLK, 0, stream>>>(bufA, counts, IN_CH, tot);
    }

    // 3) h1 = relu(agg0 @ W_rel0^T + b_rel0 + x @ W_root0^T)  -> bufB (N x 128)
    {
        int tiles = (N_NODES / 16) * (HID / 16);
        gemm_dual_relu_wmma<<<cdiv(tiles, gemmWaves), BLK, 0, stream>>>(
            bufA, W_rel0, IN_CH, x, W_root0, IN_CH, b_rel0, bufB, N_NODES, HID);
    }

    // 4) h2 = relu(h1 @ W_ro0^T + b_ro0) -> bufC
    {
        int tiles = (N_NODES / 16) * (HID / 16);
        gemm_dual_relu_wmma<<<cdiv(tiles, gemmWaves), BLK, 0, stream>>>(
            bufB, W_ro0, HID, nullptr, nullptr, 0, b_ro0, bufC, N_NODES, HID);
    }

    // 5) conv1 aggregate: agg1 (N x 128) in bufA
    {
        int tot = NM;
        zero_kernel<<<cdiv(tot, BLK), BLK, 0, stream>>>(bufA, tot);
        int etot = N_EDGES * HID;
        scatter_add_kernel<<<cdiv(etot, BLK), BLK, 0, stream>>>(src, dst, ew, bufC, bufA, HID, etot);
        scale_kernel<<<cdiv(tot, BLK), BLK, 0, stream>>>(bufA, counts, HID, tot);
    }

    // 6) h3 = relu(agg1 @ W_rel1^T + b_rel1 + h2 @ W_root1^T) -> bufB
    {
        int tiles = (N_NODES / 16) * (HID / 16);
        gemm_dual_relu_wmma<<<cdiv(tiles, gemmWaves), BLK, 0, stream>>>(
            bufA, W_rel1, HID, bufC, W_root1, HID, b_rel1, bufB, N_NODES, HID);
    }

    // 7) h4 = relu(h3 @ W_ro1^T + b_ro1) -> bufC
    {
        int tiles = (N_NODES / 16) * (HID / 16);
        gemm_dual_relu_wmma<<<cdiv(tiles, gemmWaves), BLK, 0, stream>>>(
            bufB, W_ro1, HID, nullptr, nullptr, 0, b_ro1, bufC, N_NODES, HID);
    }

    // 8) out = h4 @ W_prd^T + b_prd
    predict_kernel<<<cdiv(N_NODES, BLK), BLK, 0, stream>>>(bufC, W_prd, b_prd, out, N_NODES, HID);
}